// GCN_86938728005962
// MI455X (gfx1250) — compile-verified
//
#include <hip/hip_runtime.h>

typedef __attribute__((ext_vector_type(2))) float v2f;
typedef __attribute__((ext_vector_type(8))) float v8f;

#define NNODES 100000
#define NEDGES 3200000
#define INF    512
#define HIDF   16
#define NCLS   7

// ---------------- degree / normalization ----------------

__global__ void gcn_deg_init(float* __restrict__ deg) {
    int i = blockIdx.x * blockDim.x + threadIdx.x;
    if (i < NNODES) deg[i] = 1.0f;   // self-loop contributes 1 to every node's in-degree
}

__global__ void gcn_deg_count(const int* __restrict__ dst, float* __restrict__ deg) {
    int e = blockIdx.x * blockDim.x + threadIdx.x;
    if (e < NEDGES) atomicAdd(&deg[dst[e]], 1.0f);
}

__global__ void gcn_deg_rsqrt(float* __restrict__ deg) {
    int i = blockIdx.x * blockDim.x + threadIdx.x;
    if (i < NNODES) deg[i] = rsqrtf(deg[i]);   // deg >= 1 always (self-loops)
}

// ---------------- layer-1 GEMM: h = x @ W1 via V_WMMA_F32_16X16X4_F32 ----------------

__global__ void __launch_bounds__(256)
gcn_gemm1(const float* __restrict__ x, const float* __restrict__ W1, float* __restrict__ h) {
    __shared__ float sW[INF * HIDF];                  // 32 KB of the 320 KB WGP LDS
    for (int i = threadIdx.x; i < INF * HIDF; i += 256) sW[i] = W1[i];
    __syncthreads();

    const int wave = (blockIdx.x * 256 + threadIdx.x) >> 5;
    const int lane = threadIdx.x & 31;
    const int row0 = wave * 16;
    if (row0 >= NNODES) return;                       // wave-uniform: EXEC stays all-1s

    const int g = lane >> 4;                          // lane group (0/1)
    const int m = lane & 15;                          // row for A, col for B/D
    const float* xrow = x + (size_t)(row0 + m) * INF + 2 * g;

    v8f acc = {0.f, 0.f, 0.f, 0.f, 0.f, 0.f, 0.f, 0.f};
    #pragma unroll 8
    for (int k = 0; k < INF; k += 4) {
        float2 av = *(const float2*)(xrow + k);       // A[m][k+2g], A[m][k+2g+1]
        v2f a; a.x = av.x; a.y = av.y;
        const int kk = k + 2 * g;
        v2f b;
        b.x = sW[kk * HIDF + m];                      // B[k+2g][m]
        b.y = sW[(kk + 1) * HIDF + m];                // B[k+2g+1][m]
        acc = __builtin_amdgcn_wmma_f32_16x16x4_f32(
            false, a, false, b, (short)0, acc, false, false);
    }
    // D: VGPR r -> row (row0 + r + 8g), col m
    float* hout = h + (size_t)(row0 + 8 * g) * HIDF + m;
    #pragma unroll
    for (int r = 0; r < 8; ++r) hout[(size_t)r * HIDF] = acc[r];
}

// ---------------- layer-1 aggregation ----------------

__global__ void gcn_agg1_init(const float* __restrict__ h, const float* __restrict__ dinv,
                              const float* __restrict__ b1, float* __restrict__ agg) {
    int idx = blockIdx.x * blockDim.x + threadIdx.x;  // N*16 = 1.6M
    if (idx < NNODES * HIDF) {
        int n = idx >> 4, f = idx & 15;
        float di = dinv[n];
        agg[idx] = h[idx] * di * di + b1[f];          // self-loop term + bias
    }
}

__global__ void gcn_agg1_edges(const int* __restrict__ src, const int* __restrict__ dst,
                               const float* __restrict__ h, const float* __restrict__ dinv,
                               float* __restrict__ agg) {
    int t = blockIdx.x * blockDim.x + threadIdx.x;    // E*16 = 51.2M < 2^31
    if (t >= NEDGES * HIDF) return;
    int e = t >> 4, f = t & 15;
    int s = src[e], d = dst[e];
    float w = dinv[s] * dinv[d];
    atomicAdd(&agg[(size_t)d * HIDF + f], h[(size_t)s * HIDF + f] * w);
}

__global__ void gcn_relu(float* __restrict__ a, int n) {
    int i = blockIdx.x * blockDim.x + threadIdx.x;
    if (i < n) a[i] = fmaxf(a[i], 0.0f);
}

// ---------------- layer-2 GEMM: h2 = h1 @ W2 (7 cols, zero-padded to 16) ----------------

__global__ void __launch_bounds__(256)
gcn_gemm2(const float* __restrict__ h1, const float* __restrict__ W2, float* __restrict__ h2) {
    __shared__ float sW[HIDF * 16];
    for (int i = threadIdx.x; i < HIDF * 16; i += 256) {
        int kk = i >> 4, n = i & 15;
        sW[i] = (n < NCLS) ? W2[kk * NCLS + n] : 0.0f;
    }
    __syncthreads();

    const int wave = (blockIdx.x * 256 + threadIdx.x) >> 5;
    const int lane = threadIdx.x & 31;
    const int row0 = wave * 16;
    if (row0 >= NNODES) return;                       // wave-uniform

    const int g = lane >> 4, m = lane & 15;
    const float* hrow = h1 + (size_t)(row0 + m) * HIDF + 2 * g;

    v8f acc = {0.f, 0.f, 0.f, 0.f, 0.f, 0.f, 0.f, 0.f};
    #pragma unroll
    for (int k = 0; k < HIDF; k += 4) {
        float2 av = *(const float2*)(hrow + k);
        v2f a; a.x = av.x; a.y = av.y;
        const int kk = k + 2 * g;
        v2f b;
        b.x = sW[kk * 16 + m];
        b.y = sW[(kk + 1) * 16 + m];
        acc = __builtin_amdgcn_wmma_f32_16x16x4_f32(
            false, a, false, b, (short)0, acc, false, false);
    }
    if (m < NCLS) {                                   // predication only on stores (post-WMMA)
        float* o = h2 + (size_t)(row0 + 8 * g) * NCLS + m;
        #pragma unroll
        for (int r = 0; r < 8; ++r) o[(size_t)r * NCLS] = acc[r];
    }
}

// ---------------- layer-2 aggregation (into d_out) ----------------

__global__ void gcn_agg2_init(const float* __restrict__ h2, const float* __restrict__ dinv,
                              const float* __restrict__ b2, float* __restrict__ out) {
    int idx = blockIdx.x * blockDim.x + threadIdx.x;  // N*7 = 700000
    if (idx < NNODES * NCLS) {
        int n = idx / NCLS, c = idx - n * NCLS;
        float di = dinv[n];
        out[idx] = h2[idx] * di * di + b2[c];
    }
}

__global__ void gcn_agg2_edges(const int* __restrict__ src, const int* __restrict__ dst,
                               const float* __restrict__ h2, const float* __restrict__ dinv,
                               float* __restrict__ out) {
    int t = blockIdx.x * blockDim.x + threadIdx.x;    // E*8 = 25.6M, lane 7 of each 8 idles
    if (t >= NEDGES * 8) return;
    int e = t >> 3, c = t & 7;
    if (c >= NCLS) return;
    int s = src[e], d = dst[e];
    float w = dinv[s] * dinv[d];
    atomicAdd(&out[(size_t)d * NCLS + c], h2[(size_t)s * NCLS + c] * w);
}

// ---------------- host launch ----------------

extern "C" void kernel_launch(void* const* d_in, const int* in_sizes, int n_in,
                              void* d_out, int out_size, void* d_ws, size_t ws_size,
                              hipStream_t stream) {
    (void)in_sizes; (void)n_in; (void)out_size; (void)ws_size;
    const float* x   = (const float*)d_in[0];
    const int*   ei  = (const int*)d_in[1];     // [2, E] flat: first E = src, next E = dst
    const int*   src = ei;
    const int*   dst = ei + NEDGES;
    const float* W1  = (const float*)d_in[2];
    const float* b1  = (const float*)d_in[3];
    const float* W2  = (const float*)d_in[4];
    const float* b2  = (const float*)d_in[5];
    float*       out = (float*)d_out;

    char*  ws  = (char*)d_ws;
    size_t off = 0;
    auto carve = [&](size_t bytes) -> void* {
        void* p = ws + off;
        off = (off + bytes + 255) & ~(size_t)255;
        return p;
    };
    float* dinv = (float*)carve((size_t)NNODES * 4);               // 0.4 MB
    float* h    = (float*)carve((size_t)NNODES * HIDF * 4);        // 6.4 MB
    float* agg1 = (float*)carve((size_t)NNODES * HIDF * 4);        // 6.4 MB
    float* h2   = (float*)carve((size_t)NNODES * NCLS * 4);        // 2.8 MB

    constexpr int B = 256;
    const int nTiles     = (NNODES + 15) / 16;        // 6250 waves (wave32, 1 tile/wave)
    const int gemmBlocks = (nTiles + 7) / 8;          // 8 waves per 256-thread block

    // normalization
    gcn_deg_init <<<(NNODES + B - 1) / B, B, 0, stream>>>(dinv);
    gcn_deg_count<<<(NEDGES + B - 1) / B, B, 0, stream>>>(dst, dinv);
    gcn_deg_rsqrt<<<(NNODES + B - 1) / B, B, 0, stream>>>(dinv);

    // layer 1
    gcn_gemm1<<<gemmBlocks, B, 0, stream>>>(x, W1, h);
    gcn_agg1_init <<<((size_t)NNODES * HIDF + B - 1) / B, B, 0, stream>>>(h, dinv, b1, agg1);
    gcn_agg1_edges<<<((size_t)NEDGES * HIDF + B - 1) / B, B, 0, stream>>>(src, dst, h, dinv, agg1);
    gcn_relu<<<((size_t)NNODES * HIDF + B - 1) / B, B, 0, stream>>>(agg1, NNODES * HIDF);

    // layer 2
    gcn_gemm2<<<gemmBlocks, B, 0, stream>>>(agg1, W2, h2);
    gcn_agg2_init <<<((size_t)NNODES * NCLS + B - 1) / B, B, 0, stream>>>(h2, dinv, b2, out);
    gcn_agg2_edges<<<((size_t)NEDGES * 8 + B - 1) / B, B, 0, stream>>>(src, dst, h2, dinv, out);
}